// BSplineActivationLayer_12936441496122
// MI455X (gfx1250) — compile-verified
//
#include <hip/hip_runtime.h>
#include <hip/hip_bf16.h>
#include <math.h>

// Problem constants (from the reference)
#define B_SZ   256
#define IN_SZ  512
#define OUT_SZ 512
#define NSPL   5
#define KSEG   15
#define MU_DET 20.0f
#define SIG_DET 9.0f

// Tiling
#define TILE_O 16          // one WMMA tile of output channels per block
#define TILE_B 128         // 8 waves x 16 rows of batch per block
#define CHUNK  16          // i-chunk staged through LDS (4 WMMA k-steps)
#define SPLITK 4           // split the i reduction for occupancy
#define KRANGE (IN_SZ / SPLITK)

typedef float v2f __attribute__((ext_vector_type(2)));
typedef float v8f __attribute__((ext_vector_type(8)));

__device__ __forceinline__ float softplus_f(float v) {
    // stable: max(v,0) + log1p(exp(-|v|))
    return fmaxf(v, 0.0f) + log1pf(__expf(-fabsf(v)));
}

// ---------------------------------------------------------------------------
// Stage 1: per-(o,i) spline parameter evaluation (done ONCE, not per batch b).
// Writes 5 planes of [OUT][IN] f32 into workspace:
//   P1 = b1 * g,  P2 = b2,  P3 = b3,  P4 = b4,  P5 = b5 * g
// where g = softplus(raw_gamma) / OUT.
// ---------------------------------------------------------------------------
__global__ void __launch_bounds__(256)
spline_param_kernel(const float* __restrict__ raw_gamma,
                    const float* __restrict__ w,
                    const float* __restrict__ breaks,   // [NSPL][KSEG+1]
                    const float* __restrict__ coefs,    // [NSPL][KSEG][4]
                    float* __restrict__ P) {
    int idx = blockIdx.x * blockDim.x + threadIdx.x;
    if (idx >= OUT_SZ * IN_SZ) return;

    float wn = (fminf(fmaxf(w[idx], 5.5f), 35.5f) - MU_DET) / SIG_DET;
    float g  = softplus_f(raw_gamma[idx]) * (1.0f / (float)OUT_SZ);

    float b[NSPL];
#pragma unroll
    for (int s = 0; s < NSPL; ++s) {
        const float* brk = breaks + s * (KSEG + 1);   // uniform SGPR loads
        const float* cf  = coefs + s * KSEG * 4;
        float wcl = fminf(fmaxf(wn, brk[0]), brk[KSEG] - 1e-6f);
        // searchsorted(left) - 1  ==  count(brk[j] < wcl) - 1
        int p = 0;
#pragma unroll
        for (int j = 0; j < KSEG + 1; ++j) p += (brk[j] < wcl) ? 1 : 0;
        int seg = min(max(p - 1, 0), KSEG - 1);
        float t = wcl - brk[seg];
        const float* a = cf + seg * 4;
        b[s] = ((a[0] * t + a[1]) * t + a[2]) * t + a[3];
    }
    const int N = OUT_SZ * IN_SZ;
    P[idx]         = b[0] * g;   // b1 * gamma/OUT
    P[idx + N]     = b[1];
    P[idx + 2 * N] = b[2];
    P[idx + 3 * N] = b[3];
    P[idx + 4 * N] = b[4] * g;   // b5 * gamma/OUT  (WMMA B-matrix)
}

// ---------------------------------------------------------------------------
// Stage 3: main compute. Each wave owns a 16x16 (b,o) tile in WMMA C layout.
// Linear b5-term via V_WMMA_F32_16X16X4_F32; transcendental term via VALU
// into the same v8f accumulator; split-K partials merged with f32 atomics.
// ---------------------------------------------------------------------------
__global__ void __launch_bounds__(256)
main_kernel(const float* __restrict__ x,
            const float* __restrict__ P,
            float* __restrict__ Y) {
    __shared__ float xs[TILE_B][CHUNK + 1];          // x tile, relu applied
    __shared__ float ps[4][TILE_O][CHUNK + 1];       // b1g,b2,b3,b4 [o][i]
    __shared__ float wsB[CHUNK][TILE_O + 1];         // b5g transposed [i][o]

    const int N = OUT_SZ * IN_SZ;
    const float* P1 = P;
    const float* P2 = P + N;
    const float* P3 = P + 2 * N;
    const float* P4 = P + 3 * N;
    const float* P5 = P + 4 * N;

    const int tid   = threadIdx.x;
    const int lane  = tid & 31;          // wave32
    const int wave  = tid >> 5;          // 0..7
    const int half  = lane >> 4;         // 0/1
    const int lm    = lane & 15;
    const int oBase = blockIdx.x * TILE_O;
    const int bBase = blockIdx.y * TILE_B;
    const int iBase = blockIdx.z * KRANGE;
    const int wb    = wave * 16;         // wave's local batch-row base

    v8f acc = {0.f, 0.f, 0.f, 0.f, 0.f, 0.f, 0.f, 0.f};

    for (int ic = 0; ic < KRANGE; ic += CHUNK) {
        const int i0 = iBase + ic;

        // ---- cooperative LDS staging (coalesced 64B rows) ----
#pragma unroll
        for (int j = 0; j < (TILE_B * CHUNK) / 256; ++j) {   // 8 iters
            int l = tid + j * 256;
            int row = l >> 4, col = l & 15;
            xs[row][col] = fmaxf(x[(bBase + row) * IN_SZ + i0 + col], 0.0f);
        }
        {
            int n = tid >> 4, ik = tid & 15;
            int gidx = (oBase + n) * IN_SZ + i0 + ik;
            ps[0][n][ik] = P1[gidx];
            ps[1][n][ik] = P2[gidx];
            ps[2][n][ik] = P3[gidx];
            ps[3][n][ik] = P4[gidx];
            wsB[ik][n]   = P5[gidx];
        }
        if (ic + CHUNK < KRANGE) {   // pull next x chunk toward the caches
            __builtin_prefetch(&x[(bBase + (tid >> 4)) * IN_SZ + i0 + CHUNK], 0, 0);
        }
        __syncthreads();

        // ---- linear term: C += A(16x4,f32) * B(4x16,f32), 4 k-steps ----
#pragma unroll
        for (int kk = 0; kk < CHUNK; kk += 4) {
            v2f a, bf;
            // A layout: lanes 0-15 hold M=lane K={0,1}; lanes 16-31 K={2,3}
            a.x = xs[wb + lm][kk + 2 * half];
            a.y = xs[wb + lm][kk + 2 * half + 1];
            // B layout (4x16): rows striped across lanes, K split by half
            bf.x = wsB[kk + 2 * half][lm];
            bf.y = wsB[kk + 2 * half + 1][lm];
            acc = __builtin_amdgcn_wmma_f32_16x16x4_f32(
                false, a, false, bf, (short)0, acc, false, false);
        }

        // ---- transcendental term, accumulated in WMMA C layout ----
        //   C element (r, lane): M = 8*half + r, N = lm
#pragma unroll 4
        for (int ik = 0; ik < CHUNK; ++ik) {
            float p1 = ps[0][lm][ik];   // b1*g   (broadcast across halves)
            float p2 = ps[1][lm][ik];
            float p3 = ps[2][lm][ik];
            float p4 = ps[3][lm][ik];
#pragma unroll
            for (int r = 0; r < 8; ++r) {
                float xv   = xs[wb + 8 * half + r][ik];       // LDS broadcast
                float base = __expf(p3 * xv) - 1.0f;          // >= 0
                float bp   = __expf(p4 * __logf(base));       // base^b4
                float l1   = __logf(1.0f + bp);
                float l2   = __logf(1.0f + p2 * l1);
                acc[r] += p1 * l2;
            }
        }
        __syncthreads();
    }

    // ---- merge split-K partials ----
#pragma unroll
    for (int r = 0; r < 8; ++r) {
        int gb = bBase + wb + 8 * half + r;
        int go = oBase + lm;
        unsafeAtomicAdd(&Y[gb * OUT_SZ + go], acc[r]);   // global_atomic_add_f32
    }
}

// ---------------------------------------------------------------------------
// Stage 4: out = softplus(Y)
// ---------------------------------------------------------------------------
__global__ void __launch_bounds__(256)
finalize_kernel(const float* __restrict__ Y, float* __restrict__ out) {
    int t = blockIdx.x * blockDim.x + threadIdx.x;
    if (t < B_SZ * OUT_SZ) out[t] = softplus_f(Y[t]);
}

// ---------------------------------------------------------------------------
extern "C" void kernel_launch(void* const* d_in, const int* in_sizes, int n_in,
                              void* d_out, int out_size, void* d_ws, size_t ws_size,
                              hipStream_t stream) {
    const float* x         = (const float*)d_in[0];
    const float* raw_gamma = (const float*)d_in[1];
    const float* w         = (const float*)d_in[2];
    const float* breaks    = (const float*)d_in[3];
    const float* coefs     = (const float*)d_in[4];
    float* out = (float*)d_out;

    // ws layout: 5 param planes (5*262144 f32 = 5.24 MB) + Y acc (0.5 MB)
    float* P = (float*)d_ws;
    float* Y = P + 5 * OUT_SZ * IN_SZ;

    spline_param_kernel<<<(OUT_SZ * IN_SZ) / 256, 256, 0, stream>>>(
        raw_gamma, w, breaks, coefs, P);

    hipMemsetAsync(Y, 0, (size_t)B_SZ * OUT_SZ * sizeof(float), stream);

    dim3 grid(OUT_SZ / TILE_O, B_SZ / TILE_B, SPLITK);   // 32 x 2 x 4 = 256 blocks
    main_kernel<<<grid, 256, 0, stream>>>(x, P, Y);

    finalize_kernel<<<(B_SZ * OUT_SZ) / 256, 256, 0, stream>>>(Y, out);
}